// LoRALSTMModel_66434554134960
// MI455X (gfx1250) — compile-verified
//
#include <hip/hip_runtime.h>
#include <math.h>

typedef __attribute__((ext_vector_type(2))) float v2f;
typedef __attribute__((ext_vector_type(4))) float v4f;
typedef __attribute__((ext_vector_type(8))) float v8f;

#define TT 256
#define BB 64
#define II 1024
#define LL 2
#define GG 4
#define HH 1024
#define RR 16

// ---------------------------------------------------------------------------
// Kernel 1: LoRA down-projection  u[g,b,r] = sum_i act[b,i] * A[g,r,i]
// Tiny (2*4096 dot products of length 1024) -> plain VALU with float4 loads.
// ---------------------------------------------------------------------------
__global__ void lora_proj_kernel(const float* __restrict__ x,
                                 const float* __restrict__ h,
                                 const float* __restrict__ Ax,  // [G,R,I]
                                 const float* __restrict__ Ah,  // [G,R,H]
                                 float* __restrict__ ux,        // [G,B,R]
                                 float* __restrict__ uh) {      // [G,B,R]
    int t = blockIdx.x * blockDim.x + threadIdx.x;   // 0 .. 2*G*B*R-1
    const int NX = GG * BB * RR;
    bool isX = t < NX;
    int t2 = isX ? t : t - NX;
    int g = t2 / (BB * RR);
    int b = (t2 / RR) % BB;
    int r = t2 % RR;
    const float* act = isX ? x : h;
    const float* A = (isX ? Ax : Ah) + (size_t)(g * RR + r) * II;
    const float* v = act + (size_t)b * II;
    float acc = 0.f;
#pragma unroll 4
    for (int i = 0; i < II; i += 4) {
        v4f av = *(const v4f*)(A + i);
        v4f vv = *(const v4f*)(v + i);
        acc += av.x * vv.x + av.y * vv.y + av.z * vv.z + av.w * vv.w;
    }
    (isX ? ux : uh)[t2] = acc;
}

// ---------------------------------------------------------------------------
// Kernel 2: pre[g,b,n] = x·Wx[g]^T + h·Wh[g]^T + ux·Bx[g]^T + uh·Bh[g]^T
// grid.x = G * (H/32), block = 128 threads (4 waves, wave32).
// Wave w owns batch rows m = 16w..16w+15; each wave holds two 16x16 f32
// accumulators (columns nbase..nbase+15 and nbase+16..nbase+31).
// V_WMMA_F32_16X16X4_F32: A lane(lm)=m, half(lh) selects K pair; B lane=n.
// ---------------------------------------------------------------------------
__global__ __launch_bounds__(128)
void lstm_gemm_kernel(const float* __restrict__ x,    // [B,I]
                      const float* __restrict__ h,    // [B,H]
                      const float* __restrict__ Wx,   // [G,H,I] (layer base)
                      const float* __restrict__ Wh,   // [G,H,H]
                      const float* __restrict__ ux,   // [G,B,R]
                      const float* __restrict__ uh,   // [G,B,R]
                      const float* __restrict__ Bx,   // [G,H,R]
                      const float* __restrict__ Bh,   // [G,H,R]
                      float* __restrict__ pre) {      // [G,B,H]
    const int wave = threadIdx.x >> 5;
    const int lane = threadIdx.x & 31;
    const int lm = lane & 15;   // m (A) / n (B) within tile
    const int lh = lane >> 4;   // selects K pair within K=4 step
    const int g = blockIdx.x >> 5;
    const int nbase = (blockIdx.x & 31) * 32;
    const int m = wave * 16 + lm;
    const int n0 = nbase + lm;
    const int n1 = n0 + 16;
    const int koff = 2 * lh;

    v8f c0 = {};
    v8f c1 = {};

    // ---- x @ Wx^T (K = I = 1024) ----
    {
        const float* arow = x + (size_t)m * II;
        const float* w0 = Wx + ((size_t)g * HH + n0) * II;
        const float* w1 = Wx + ((size_t)g * HH + n1) * II;
#pragma unroll 4
        for (int k = 0; k < II; k += 4) {
            int kk = k + koff;
            v2f a  = *(const v2f*)(arow + kk);
            v2f b0 = *(const v2f*)(w0 + kk);
            v2f b1 = *(const v2f*)(w1 + kk);
            c0 = __builtin_amdgcn_wmma_f32_16x16x4_f32(false, a, false, b0,
                                                       (short)0, c0, false, false);
            c1 = __builtin_amdgcn_wmma_f32_16x16x4_f32(false, a, false, b1,
                                                       (short)0, c1, false, false);
        }
    }

    // ---- h @ Wh^T (K = H = 1024) ----
    {
        const float* arow = h + (size_t)m * HH;
        const float* w0 = Wh + ((size_t)g * HH + n0) * HH;
        const float* w1 = Wh + ((size_t)g * HH + n1) * HH;
#pragma unroll 4
        for (int k = 0; k < HH; k += 4) {
            int kk = k + koff;
            v2f a  = *(const v2f*)(arow + kk);
            v2f b0 = *(const v2f*)(w0 + kk);
            v2f b1 = *(const v2f*)(w1 + kk);
            c0 = __builtin_amdgcn_wmma_f32_16x16x4_f32(false, a, false, b0,
                                                       (short)0, c0, false, false);
            c1 = __builtin_amdgcn_wmma_f32_16x16x4_f32(false, a, false, b1,
                                                       (short)0, c1, false, false);
        }
    }

    // ---- LoRA up-projection: ux @ Bx^T (K = R = 16) ----
    {
        const float* arow = ux + ((size_t)g * BB + m) * RR;
        const float* w0 = Bx + ((size_t)g * HH + n0) * RR;
        const float* w1 = Bx + ((size_t)g * HH + n1) * RR;
#pragma unroll
        for (int k = 0; k < RR; k += 4) {
            int kk = k + koff;
            v2f a  = *(const v2f*)(arow + kk);
            v2f b0 = *(const v2f*)(w0 + kk);
            v2f b1 = *(const v2f*)(w1 + kk);
            c0 = __builtin_amdgcn_wmma_f32_16x16x4_f32(false, a, false, b0,
                                                       (short)0, c0, false, false);
            c1 = __builtin_amdgcn_wmma_f32_16x16x4_f32(false, a, false, b1,
                                                       (short)0, c1, false, false);
        }
    }

    // ---- LoRA up-projection: uh @ Bh^T (K = R = 16) ----
    {
        const float* arow = uh + ((size_t)g * BB + m) * RR;
        const float* w0 = Bh + ((size_t)g * HH + n0) * RR;
        const float* w1 = Bh + ((size_t)g * HH + n1) * RR;
#pragma unroll
        for (int k = 0; k < RR; k += 4) {
            int kk = k + koff;
            v2f a  = *(const v2f*)(arow + kk);
            v2f b0 = *(const v2f*)(w0 + kk);
            v2f b1 = *(const v2f*)(w1 + kk);
            c0 = __builtin_amdgcn_wmma_f32_16x16x4_f32(false, a, false, b0,
                                                       (short)0, c0, false, false);
            c1 = __builtin_amdgcn_wmma_f32_16x16x4_f32(false, a, false, b1,
                                                       (short)0, c1, false, false);
        }
    }

    // ---- store D: VGPR j holds m = 16*wave + 8*lh + j, n = lane%16 ----
    float* outg = pre + (size_t)g * BB * HH;
#pragma unroll
    for (int j = 0; j < 8; ++j) {
        int mm = wave * 16 + lh * 8 + j;
        outg[(size_t)mm * HH + n0] = c0[j];
        outg[(size_t)mm * HH + n1] = c1[j];
    }
}

// ---------------------------------------------------------------------------
// Kernel 3: LSTM gating (adds biases), writes h_t[l], c_t[l] (+ out for last
// layer). 64K elements -> trivially bandwidth bound.
// ---------------------------------------------------------------------------
__global__ void lstm_gate_kernel(const float* __restrict__ pre,  // [G,B,H]
                                 const float* __restrict__ cin,  // [B,H]
                                 const float* __restrict__ bx,   // [G,H]
                                 const float* __restrict__ bh,   // [G,H]
                                 float* __restrict__ h_out,
                                 float* __restrict__ c_out,
                                 float* __restrict__ out_dup) {
    int t = blockIdx.x * blockDim.x + threadIdx.x;  // 0 .. B*H-1
    int hcol = t & (HH - 1);
    float p0 = pre[t]               + bx[0 * HH + hcol] + bh[0 * HH + hcol];
    float p1 = pre[t + 1 * BB * HH] + bx[1 * HH + hcol] + bh[1 * HH + hcol];
    float p2 = pre[t + 2 * BB * HH] + bx[2 * HH + hcol] + bh[2 * HH + hcol];
    float p3 = pre[t + 3 * BB * HH] + bx[3 * HH + hcol] + bh[3 * HH + hcol];
    float it = 1.f / (1.f + __expf(-p0));
    float ft = 1.f / (1.f + __expf(-p1));
    float gt = tanhf(p2);
    float ot = 1.f / (1.f + __expf(-p3));
    float cn = ft * cin[t] + it * gt;
    float hn = ot * tanhf(cn);
    h_out[t] = hn;
    c_out[t] = cn;
    if (out_dup) out_dup[t] = hn;
}

// ---------------------------------------------------------------------------
extern "C" void kernel_launch(void* const* d_in, const int* in_sizes, int n_in,
                              void* d_out, int out_size, void* d_ws, size_t ws_size,
                              hipStream_t stream) {
    (void)in_sizes; (void)n_in; (void)out_size; (void)ws_size;
    const float* input_seq = (const float*)d_in[0];  // [T,B,I]
    const float* h0  = (const float*)d_in[1];        // [L,B,H]
    const float* c0  = (const float*)d_in[2];        // [L,B,H]
    const float* W_x = (const float*)d_in[3];        // [L,G,H,I]
    const float* W_h = (const float*)d_in[4];        // [L,G,H,H]
    const float* b_x = (const float*)d_in[5];        // [L,G,H]
    const float* b_h = (const float*)d_in[6];        // [L,G,H]
    const float* A_x = (const float*)d_in[7];        // [L,G,R,I]
    const float* B_x = (const float*)d_in[8];        // [L,G,H,R]
    const float* A_h = (const float*)d_in[9];        // [L,G,R,H]
    const float* B_h = (const float*)d_in[10];       // [L,G,H,R]
    float* out = (float*)d_out;                      // [B,H] + [L,B,H] + [L,B,H]

    float* pre = (float*)d_ws;                       // G*B*H floats
    float* ux  = pre + (size_t)GG * BB * HH;         // G*B*R
    float* uh  = ux + GG * BB * RR;                  // G*B*R

    float* h_t = out + (size_t)BB * HH;
    float* c_t = out + (size_t)BB * HH + (size_t)LL * BB * HH;

    const float* xl = input_seq + (size_t)(TT - 1) * BB * II;  // last timestep
    for (int l = 0; l < LL; ++l) {
        const float* hl  = h0 + (size_t)l * BB * HH;
        const float* cl  = c0 + (size_t)l * BB * HH;
        const float* Wxl = W_x + (size_t)l * GG * HH * II;
        const float* Whl = W_h + (size_t)l * GG * HH * HH;
        const float* Axl = A_x + (size_t)l * GG * RR * II;
        const float* Bxl = B_x + (size_t)l * GG * HH * RR;
        const float* Ahl = A_h + (size_t)l * GG * RR * HH;
        const float* Bhl = B_h + (size_t)l * GG * HH * RR;
        const float* bxl = b_x + (size_t)l * GG * HH;
        const float* bhl = b_h + (size_t)l * GG * HH;
        float* hout = h_t + (size_t)l * BB * HH;
        float* cout_ = c_t + (size_t)l * BB * HH;

        hipLaunchKernelGGL(lora_proj_kernel, dim3(2 * GG * BB * RR / 256),
                           dim3(256), 0, stream, xl, hl, Axl, Ahl, ux, uh);
        hipLaunchKernelGGL(lstm_gemm_kernel, dim3(GG * (HH / 32)),
                           dim3(128), 0, stream,
                           xl, hl, Wxl, Whl, ux, uh, Bxl, Bhl, pre);
        hipLaunchKernelGGL(lstm_gate_kernel, dim3(BB * HH / 256),
                           dim3(256), 0, stream,
                           pre, cl, bxl, bhl, hout, cout_,
                           (l == LL - 1) ? out : nullptr);
        xl = hout;  // next layer's input is this layer's hidden state
    }
}